// DYANrhEnc_23244363006308
// MI455X (gfx1250) — compile-verified
//
#include <hip/hip_runtime.h>
#include <math.h>

// ---------------- problem constants (match reference) ----------------
#define N_   256
#define T_   36
#define P_   80
#define K_   161            // 1 + 2*NPOLE
#define F_   50
#define M_   (N_ * F_)      // 12800 independent FISTA columns
#define KP   176            // K padded to 11 * 16 for WMMA tiles
#define KT   (KP / 4)       // 44 k-steps of V_WMMA_F32_16X16X4_F32
#define MT   (KP / 16)      // 11 row tiles
#define COLS 64             // columns per block (4 waves x 16 cols)
#define FISTA_IT 100
#define LAM  0.1f

typedef float v2f __attribute__((ext_vector_type(2)));
typedef float v4f __attribute__((ext_vector_type(4)));
typedef float v8f __attribute__((ext_vector_type(8)));

// ---------------- workspace layout (floats) ----------------
static constexpr size_t WS_D     = 0;                       // T_*K_       = 5796
static constexpr size_t WS_DTD   = WS_D + (size_t)T_ * K_;  // K_*K_       = 25921
static constexpr size_t WS_SC    = WS_DTD + (size_t)K_ * K_;// 8 scalars
static constexpr size_t WS_A     = ((WS_SC + 8 + 63) / 64) * 64;        // KP*KP
static constexpr size_t WS_DTYL  = WS_A + (size_t)KP * KP;              // KP*M_
static constexpr size_t WS_WL    = WS_DTYL + (size_t)KP * M_;           // KP*M_
static constexpr size_t WS_CODE  = WS_WL + (size_t)KP * M_;             // KP*M_
// total ~6.83M floats ~= 27.4 MB

// sc[0]=frob accum, sc[1]=Linv, sc[2]=wn norm accum, sc[3]=wl scale

// ---------------- output layout (floats, return order: code, D, reconst) ----
static constexpr size_t OUT_CODE = 0;
static constexpr size_t OUT_D    = (size_t)N_ * K_ * F_;    // 2,060,800
static constexpr size_t OUT_R    = OUT_D + (size_t)T_ * K_; // +5,796
// total = 2,527,396

// =====================================================================
__global__ void zero_scalars_kernel(float* sc) {
    if (threadIdx.x < 8) sc[threadIdx.x] = 0.0f;
}

// D[t,k] = [1 | rr^t cos(t*theta) | rr^t sin(t*theta)]
__global__ void build_D_kernel(const float* __restrict__ rr,
                               const float* __restrict__ theta,
                               float* __restrict__ dD,
                               float* __restrict__ out_D) {
    int idx = blockIdx.x * blockDim.x + threadIdx.x;
    if (idx >= T_ * K_) return;
    int t = idx / K_, k = idx % K_;
    float v;
    if (k == 0) {
        v = 1.0f;
    } else if (k <= P_) {
        float r = rr[k - 1], th = theta[k - 1];
        v = powf(r, (float)t) * cosf((float)t * th);
    } else {
        float r = rr[k - 1 - P_], th = theta[k - 1 - P_];
        v = powf(r, (float)t) * sinf((float)t * th);
    }
    dD[idx]    = v;
    out_D[idx] = v;
}

__global__ void dtd_kernel(const float* __restrict__ dD, float* __restrict__ dtd) {
    int idx = blockIdx.x * blockDim.x + threadIdx.x;
    if (idx >= K_ * K_) return;
    int i = idx / K_, j = idx % K_;
    float s = 0.0f;
    for (int t = 0; t < T_; ++t) s = fmaf(dD[t * K_ + i], dD[t * K_ + j], s);
    dtd[idx] = s;
}

__global__ void frob_kernel(const float* __restrict__ dtd, float* sc) {
    __shared__ float red[256];
    int idx = blockIdx.x * blockDim.x + threadIdx.x;
    float s = 0.0f;
    if (idx < K_ * K_) { float v = dtd[idx]; s = v * v; }
    red[threadIdx.x] = s;
    __syncthreads();
    for (int d = 128; d > 0; d >>= 1) {
        if ((int)threadIdx.x < d) red[threadIdx.x] += red[threadIdx.x + d];
        __syncthreads();
    }
    if (threadIdx.x == 0) atomicAdd(&sc[0], red[0]);
}

__global__ void finalize_L_kernel(float* sc) {
    sc[1] = rsqrtf(sc[0]);          // Linv = 1 / ||DtD||_F
}

// A = I - DtD * Linv, zero-padded to KP x KP (row-major)
__global__ void build_A_kernel(const float* __restrict__ dtd,
                               const float* __restrict__ sc,
                               float* __restrict__ A_p) {
    int idx = blockIdx.x * blockDim.x + threadIdx.x;
    if (idx >= KP * KP) return;
    int row = idx / KP, col = idx % KP;
    float v = 0.0f;
    if (row < K_ && col < K_)
        v = ((row == col) ? 1.0f : 0.0f) - dtd[row * K_ + col] * sc[1];
    A_p[idx] = v;
}

// dtyL[col*KP + k] = Linv * sum_t D[t,k] * x[n,t,f], col = n*F + f (padded rows 0)
__global__ void dtyL_kernel(const float* __restrict__ x,
                            const float* __restrict__ dD,
                            const float* __restrict__ sc,
                            float* __restrict__ dtyL) {
    size_t idx = (size_t)blockIdx.x * blockDim.x + threadIdx.x;
    if (idx >= (size_t)KP * M_) return;
    int col = (int)(idx / KP), k = (int)(idx % KP);
    float v = 0.0f;
    if (k < K_) {
        int n = col / F_, f = col % F_;
        const float* xc = x + ((size_t)n * T_) * F_ + f;
        float s = 0.0f;
        for (int t = 0; t < T_; ++t) s = fmaf(dD[t * K_ + k], xc[(size_t)t * F_], s);
        v = s * sc[1];
    }
    dtyL[idx] = v;
}

// round 1: w == 1  ->  wl = lambda * Linv everywhere valid
__global__ void wl_init_kernel(const float* __restrict__ sc, float* __restrict__ wl) {
    size_t idx = (size_t)blockIdx.x * blockDim.x + threadIdx.x;
    if (idx >= (size_t)KP * M_) return;
    int k = (int)(idx % KP);
    wl[idx] = (k < K_) ? LAM * sc[1] : 0.0f;
}

// =====================================================================
// Core: 100 FISTA iterations, fully resident. 128 threads = 4 waves;
// each wave owns 16 columns. A (176x176) + y/c/wl slabs live in LDS
// (~253 KB of the 320 KB WGP budget, via dynamic LDS). x state lives in
// VGPRs in WMMA C/D layout. No barriers in the iteration loop: y is
// wave-private and same-wave DS ops are in-order (DScnt rules).
__global__ void __launch_bounds__(128, 1)
fista_kernel(const float* __restrict__ A_g,
             const float* __restrict__ c_g,   // DtY * Linv  (KP x M, col-major)
             const float* __restrict__ w_g,   // wl          (KP x M, col-major)
             float* __restrict__ codebuf) {   // out x       (KP x M, col-major)
    extern __shared__ float smem[];
    float* A_s = smem;                         // KP*KP, row-major [row][k]
    float* y_s = A_s + (size_t)KP * KP;        // KP*COLS, col-major [col][k]
    float* c_s = y_s + (size_t)KP * COLS;
    float* w_s = c_s + (size_t)KP * COLS;

    const int tid  = threadIdx.x;
    const int lane = tid & 31;
    const int wave = tid >> 5;
    const int nlo  = lane & 15;
    const int hi   = lane >> 4;
    const size_t blkcol = (size_t)blockIdx.x * COLS;

    for (int i = tid; i < KP * KP; i += 128) A_s[i] = A_g[i];
    for (int i = tid; i < KP * COLS; i += 128) {
        c_s[i] = c_g[blkcol * KP + i];
        w_s[i] = w_g[blkcol * KP + i];
        y_s[i] = 0.0f;                         // y0 = x0 = 0
    }
    __syncthreads();

    const int col    = wave * 16 + nlo;        // this lane's column (0..63)
    const int aoff   = nlo * KP + 2 * hi;      // + m*16*KP + kt*4   (A operand)
    const int boff   = col * KP + 2 * hi;      // + kt*4             (B operand)
    const int cdbase = col * KP + 8 * hi;      // + m*16             (C/D rows)

    v8f x[MT];
#pragma unroll
    for (int m = 0; m < MT; ++m) x[m] = (v8f){0,0,0,0,0,0,0,0};

    float tmom = 1.0f;
#pragma unroll 1
    for (int it = 0; it < FISTA_IT; ++it) {
        v8f acc[MT];
#pragma unroll
        for (int m = 0; m < MT; ++m) acc[m] = (v8f){0,0,0,0,0,0,0,0};

        // Ay = A @ y : 44 k-steps x 11 independent row-tile accumulators
#pragma unroll 4
        for (int kt = 0; kt < KT; ++kt) {
            v2f b = *(const v2f*)(y_s + boff + kt * 4);
#pragma unroll
            for (int m = 0; m < MT; ++m) {
                v2f a = *(const v2f*)(A_s + aoff + m * 16 * KP + kt * 4);
                acc[m] = __builtin_amdgcn_wmma_f32_16x16x4_f32(
                    false, a, false, b, (short)0, acc[m], false, false);
            }
        }

        float tnext = 0.5f * (1.0f + sqrtf(fmaf(4.0f * tmom, tmom, 1.0f)));
        float mom   = (tmom - 1.0f) / tnext;
        tmom = tnext;

        // shrink:  x = max(0, Ay + c - wl) + min(0, Ay + c + wl);  y = x + mom*(x - x_old)
#pragma unroll
        for (int m = 0; m < MT; ++m) {
            const int base = cdbase + m * 16;
            v4f c0 = *(const v4f*)(c_s + base);
            v4f c1 = *(const v4f*)(c_s + base + 4);
            v4f w0 = *(const v4f*)(w_s + base);
            v4f w1 = *(const v4f*)(w_s + base + 4);
            v4f ya, yb;
#pragma unroll
            for (int r = 0; r < 4; ++r) {
                float ay = acc[m][r];
                float xv = fmaxf(0.0f, ay + c0[r] - w0[r]) + fminf(0.0f, ay + c0[r] + w0[r]);
                ya[r] = fmaf(mom, xv - x[m][r], xv);
                x[m][r] = xv;
            }
#pragma unroll
            for (int r = 0; r < 4; ++r) {
                float ay = acc[m][r + 4];
                float xv = fmaxf(0.0f, ay + c1[r] - w1[r]) + fminf(0.0f, ay + c1[r] + w1[r]);
                yb[r] = fmaf(mom, xv - x[m][r + 4], xv);
                x[m][r + 4] = xv;
            }
            *(v4f*)(y_s + base)     = ya;
            *(v4f*)(y_s + base + 4) = yb;
        }
    }

    // spill resident x to global (col-major KP x M)
#pragma unroll
    for (int m = 0; m < MT; ++m) {
        size_t g = (blkcol + (size_t)col) * KP + 8 * hi + m * 16;
        v4f o0, o1;
#pragma unroll
        for (int r = 0; r < 4; ++r) { o0[r] = x[m][r]; o1[r] = x[m][r + 4]; }
        *(v4f*)(codebuf + g)     = o0;
        *(v4f*)(codebuf + g + 4) = o1;
    }
}

// =====================================================================
// reweighting:  wn = 1/(|code|+0.01);  w = wn/||wn|| * K;  wl = w * lam * Linv
__global__ void wnorm_kernel(const float* __restrict__ codebuf, float* sc) {
    __shared__ float red[256];
    int col = blockIdx.x * blockDim.x + threadIdx.x;   // 50*256 == M_
    float s = 0.0f;
    if (col < M_) {
        const float* c = codebuf + (size_t)col * KP;
        for (int k = 0; k < K_; ++k) {
            float w = 1.0f / (fabsf(c[k]) + 0.01f);
            s = fmaf(w, w, s);
        }
    }
    red[threadIdx.x] = s;
    __syncthreads();
    for (int d = 128; d > 0; d >>= 1) {
        if ((int)threadIdx.x < d) red[threadIdx.x] += red[threadIdx.x + d];
        __syncthreads();
    }
    if (threadIdx.x == 0) atomicAdd(&sc[2], red[0]);
}

__global__ void finalize_wl_kernel(float* sc) {
    sc[3] = LAM * sc[1] * (float)K_ * rsqrtf(sc[2]);
}

__global__ void wl_update_kernel(const float* __restrict__ codebuf,
                                 const float* __restrict__ sc,
                                 float* __restrict__ wl) {
    size_t idx = (size_t)blockIdx.x * blockDim.x + threadIdx.x;
    if (idx >= (size_t)KP * M_) return;
    int k = (int)(idx % KP);
    wl[idx] = (k < K_) ? sc[3] / (fabsf(codebuf[idx]) + 0.01f) : 0.0f;
}

// code (KP x M col-major)  ->  (N, K, F)
__global__ void code_out_kernel(const float* __restrict__ codebuf,
                                float* __restrict__ out) {
    size_t idx = (size_t)blockIdx.x * blockDim.x + threadIdx.x;
    if (idx >= (size_t)N_ * K_ * F_) return;
    int n = (int)(idx / ((size_t)K_ * F_));
    int r = (int)(idx % ((size_t)K_ * F_));
    int k = r / F_, f = r % F_;
    out[OUT_CODE + idx] = codebuf[(size_t)(n * F_ + f) * KP + k];
}

// reconst[n,t,f] = sum_k D[t,k] * code[n,k,f]
__global__ void reconst_kernel(const float* __restrict__ dD,
                               const float* __restrict__ codebuf,
                               float* __restrict__ out) {
    size_t idx = (size_t)blockIdx.x * blockDim.x + threadIdx.x;
    if (idx >= (size_t)N_ * T_ * F_) return;
    int n = (int)(idx / ((size_t)T_ * F_));
    int r = (int)(idx % ((size_t)T_ * F_));
    int t = r / F_, f = r % F_;
    const float* c = codebuf + (size_t)(n * F_ + f) * KP;
    const float* d = dD + (size_t)t * K_;
    float s = 0.0f;
    for (int k = 0; k < K_; ++k) s = fmaf(d[k], c[k], s);
    out[OUT_R + idx] = s;
}

// =====================================================================
extern "C" void kernel_launch(void* const* d_in, const int* in_sizes, int n_in,
                              void* d_out, int out_size, void* d_ws, size_t ws_size,
                              hipStream_t stream) {
    (void)in_sizes; (void)n_in; (void)out_size; (void)ws_size;
    const float* x     = (const float*)d_in[0];
    const float* rr    = (const float*)d_in[1];
    const float* theta = (const float*)d_in[2];
    float* out = (float*)d_out;
    float* ws  = (float*)d_ws;

    float* dD    = ws + WS_D;
    float* dtd   = ws + WS_DTD;
    float* sc    = ws + WS_SC;
    float* A_p   = ws + WS_A;
    float* dtyL  = ws + WS_DTYL;
    float* wl    = ws + WS_WL;
    float* code  = ws + WS_CODE;

    const int SMEM = (int)((size_t)(KP * KP + 3 * KP * COLS) * sizeof(float)); // 259072 B
    (void)hipFuncSetAttribute(reinterpret_cast<const void*>(fista_kernel),
                              hipFuncAttributeMaxDynamicSharedMemorySize, SMEM);

    const size_t KPM = (size_t)KP * M_;

    zero_scalars_kernel<<<1, 32, 0, stream>>>(sc);
    build_D_kernel<<<(T_ * K_ + 255) / 256, 256, 0, stream>>>(rr, theta, dD, out + OUT_D);
    dtd_kernel<<<(K_ * K_ + 255) / 256, 256, 0, stream>>>(dD, dtd);
    frob_kernel<<<(K_ * K_ + 255) / 256, 256, 0, stream>>>(dtd, sc);
    finalize_L_kernel<<<1, 1, 0, stream>>>(sc);
    build_A_kernel<<<(KP * KP) / 256, 256, 0, stream>>>(dtd, sc, A_p);
    dtyL_kernel<<<(int)((KPM + 255) / 256), 256, 0, stream>>>(x, dD, sc, dtyL);
    wl_init_kernel<<<(int)((KPM + 255) / 256), 256, 0, stream>>>(sc, wl);

    // outer reweighting round 1
    fista_kernel<<<M_ / COLS, 128, SMEM, stream>>>(A_p, dtyL, wl, code);

    // w update
    wnorm_kernel<<<M_ / 256, 256, 0, stream>>>(code, sc);
    finalize_wl_kernel<<<1, 1, 0, stream>>>(sc);
    wl_update_kernel<<<(int)((KPM + 255) / 256), 256, 0, stream>>>(code, sc, wl);

    // outer reweighting round 2
    fista_kernel<<<M_ / COLS, 128, SMEM, stream>>>(A_p, dtyL, wl, code);

    // outputs
    code_out_kernel<<<(int)(((size_t)N_ * K_ * F_ + 255) / 256), 256, 0, stream>>>(code, out);
    reconst_kernel<<<(int)(((size_t)N_ * T_ * F_ + 255) / 256), 256, 0, stream>>>(dD, code, out);
}